// DCT2D_481036337606
// MI455X (gfx1250) — compile-verified
//
#include <hip/hip_runtime.h>

// ---------------------------------------------------------------------------
// 2D 8x8 DCT as one flat GEMM:  OUT(NB x 64) = (X - 128)(NB x 64) * D(64 x 64)
// then per-column scale.  D == dct_tensor viewed as [64][64].
// Matrix engine: V_WMMA_F32_16X16X4_F32, K=64 chained as 16 chunks of 4.
// ---------------------------------------------------------------------------

typedef __attribute__((ext_vector_type(2))) float v2f;
typedef __attribute__((ext_vector_type(4))) float v4f;
typedef __attribute__((ext_vector_type(8))) float v8f;

#define WAVES_PER_BLOCK 8
#define LDS_A_STRIDE    66   // 64 + 2 dword pad to break bank conflicts

// DS ops from one wave complete in order; this fence makes cross-lane LDS
// data produced by this wave visible before the next LDS consumer phase,
// and stops the compiler from reordering across it.
#define LDS_WAVE_FENCE() asm volatile("s_wait_dscnt 0x0" ::: "memory")

__global__ __launch_bounds__(256) void dct2d_wmma_kernel(
    const float* __restrict__ x,      // [NB][64]  (NB = nblocks*channels)
    const float* __restrict__ dct,    // [64][64]  dct_tensor flattened (k=x*8+y, col=u*8+v)
    const float* __restrict__ scale,  // [64]
    float* __restrict__ out,          // [NB][64]
    int nblk, int ntiles)
{
    // B fragments pre-swizzled for the 16x16x4 B layout:
    //   entry p = chunk*4 + nt, per-lane float2:
    //   v0 = D[k0 + (lane>=16 ? 2:0)    ][col], v1 = D[k0+1+(lane>=16?2:0)][col]
    __shared__ float ldsB[16 * 4 * 32 * 2];                       // 16 KB
    __shared__ float ldsA[WAVES_PER_BLOCK][16 * LDS_A_STRIDE];    // per-wave stage

    const int tid  = threadIdx.x;
    const int lane = tid & 31;
    const int wave = tid >> 5;

    // ---- build B fragment table cooperatively (constant, once per block) ----
    for (int e = tid; e < 64 * 32; e += 256) {
        int p     = e >> 5;          // chunk*4 + nt
        int l     = e & 31;
        int chunk = p >> 2;
        int nt    = p & 3;
        int col   = nt * 16 + (l & 15);
        int krow  = chunk * 4 + ((l >> 4) << 1);   // +2 rows for upper half-wave
        ldsB[e * 2 + 0] = dct[(krow + 0) * 64 + col];
        ldsB[e * 2 + 1] = dct[(krow + 1) * 64 + col];
    }
    __syncthreads();

    // ---- per-lane column scales (column-constant in C/D layout) ----
    const int m  = lane & 15;
    const float s0 = scale[ 0 + m];
    const float s1 = scale[16 + m];
    const float s2 = scale[32 + m];
    const float s3 = scale[48 + m];

    float* A = &ldsA[wave][0];
    const int half = lane >> 4;           // 0: lanes 0-15, 1: lanes 16-31
    const int koff = half * 2;            // A layout K offset for upper half

    const int waveId     = blockIdx.x * WAVES_PER_BLOCK + wave;
    const int waveStride = gridDim.x * WAVES_PER_BLOCK;

    for (int tile = waveId; tile < ntiles; tile += waveStride) {
        const long tileBlk  = (long)tile * 16;
        const bool fullTile = (tileBlk + 16 <= (long)nblk);   // wave-uniform

        // ---- stage 16 rows (blocks): batch all 8 b128 loads, then -128 + LDS ----
        v4f vbuf[8];
        if (fullTile) {
            const float* base = x + tileBlk * 64 + half * 64 + (long)m * 4;
#pragma unroll
            for (int j = 0; j < 8; ++j)
                vbuf[j] = *(const v4f*)(base + j * 128);
        } else {
#pragma unroll
            for (int j = 0; j < 8; ++j) {
                long row  = tileBlk + 2 * j + half;
                long gofs = (row < nblk) ? (row * 64 + (long)m * 4) : 0;
                vbuf[j] = *(const v4f*)(x + gofs);
            }
        }
#pragma unroll
        for (int j = 0; j < 8; ++j) {
            v4f v = vbuf[j] - 128.0f;
            *(v4f*)(A + (2 * j + half) * LDS_A_STRIDE + m * 4) = v;
        }
        LDS_WAVE_FENCE();

        // ---- 64x WMMA f32 16x16x4: 16 K-chunks x 4 N-tiles ----
        v8f acc0 = {}, acc1 = {}, acc2 = {}, acc3 = {};
#pragma unroll
        for (int chunk = 0; chunk < 16; ++chunk) {
            v2f a  = *(const v2f*)(A + m * LDS_A_STRIDE + chunk * 4 + koff);
            v2f b0 = *(const v2f*)(ldsB + (((chunk * 4 + 0) * 32 + lane) << 1));
            v2f b1 = *(const v2f*)(ldsB + (((chunk * 4 + 1) * 32 + lane) << 1));
            v2f b2 = *(const v2f*)(ldsB + (((chunk * 4 + 2) * 32 + lane) << 1));
            v2f b3 = *(const v2f*)(ldsB + (((chunk * 4 + 3) * 32 + lane) << 1));
            acc0 = __builtin_amdgcn_wmma_f32_16x16x4_f32(false, a, false, b0, (short)0, acc0, false, false);
            acc1 = __builtin_amdgcn_wmma_f32_16x16x4_f32(false, a, false, b1, (short)0, acc1, false, false);
            acc2 = __builtin_amdgcn_wmma_f32_16x16x4_f32(false, a, false, b2, (short)0, acc2, false, false);
            acc3 = __builtin_amdgcn_wmma_f32_16x16x4_f32(false, a, false, b3, (short)0, acc3, false, false);
        }

        // ---- epilogue: per-column scale ----
        acc0 *= s0; acc1 *= s1; acc2 *= s2; acc3 *= s3;

        // ---- D (C/D layout) -> LDS stage (conflict-free b32 stores) ----
        const int rowadd = half * 8;
#pragma unroll
        for (int r = 0; r < 8; ++r) {
            float* dst = A + (r + rowadd) * LDS_A_STRIDE + m;
            dst[ 0] = acc0[r];
            dst[16] = acc1[r];
            dst[32] = acc2[r];
            dst[48] = acc3[r];
        }
        LDS_WAVE_FENCE();

        // ---- LDS -> global, coalesced b128 stores ----
        if (fullTile) {
            float* obase = out + tileBlk * 64 + half * 64 + (long)m * 4;
#pragma unroll
            for (int j = 0; j < 8; ++j) {
                v4f v = *(const v4f*)(A + (2 * j + half) * LDS_A_STRIDE + m * 4);
                *(v4f*)(obase + j * 128) = v;
            }
        } else {
#pragma unroll
            for (int j = 0; j < 8; ++j) {
                long row = tileBlk + 2 * j + half;
                if (row < nblk) {
                    v4f v = *(const v4f*)(A + (2 * j + half) * LDS_A_STRIDE + m * 4);
                    *(v4f*)(out + row * 64 + (long)m * 4) = v;
                }
            }
        }
        LDS_WAVE_FENCE();  // reads drained before next iteration overwrites stage
    }
}

extern "C" void kernel_launch(void* const* d_in, const int* in_sizes, int n_in,
                              void* d_out, int out_size, void* d_ws, size_t ws_size,
                              hipStream_t stream) {
    (void)n_in; (void)out_size; (void)d_ws; (void)ws_size;
    const float* x     = (const float*)d_in[0];   // [262144,3,8,8] fp32
    const float* dct   = (const float*)d_in[1];   // [8,8,8,8] fp32 == [64][64]
    const float* scale = (const float*)d_in[2];   // [8,8] fp32 == [64]
    float*       out   = (float*)d_out;

    const int nblk   = in_sizes[0] / 64;          // 786432 8x8 blocks
    const int ntiles = (nblk + 15) / 16;          // 49152 WMMA row-tiles

    int grid = (ntiles + WAVES_PER_BLOCK - 1) / WAVES_PER_BLOCK;
    if (grid > 6144) grid = 6144;                 // 6144*8 waves = 1 tile/wave at full size
    if (grid < 1)    grid = 1;

    dct2d_wmma_kernel<<<grid, 256, 0, stream>>>(x, dct, scale, out, nblk, ntiles);
}